// GraphAttentionLayer_3083786518812
// MI455X (gfx1250) — compile-verified
//
#include <hip/hip_runtime.h>

// ---------------- feature detection ----------------------------------------
#if __has_builtin(__builtin_amdgcn_wmma_f32_16x16x4_f32)
#define HAVE_WMMA_F32K4 1
#else
#define HAVE_WMMA_F32K4 0
#endif

#if __has_builtin(__builtin_amdgcn_global_load_async_to_lds_b128) && \
    __has_builtin(__builtin_amdgcn_global_load_async_to_lds_b32) &&  \
    __has_builtin(__builtin_amdgcn_s_wait_asynccnt)
#define HAVE_ASYNC 1
#pragma message("PROBE: HAS async_to_lds path")
#else
#define HAVE_ASYNC 0
#pragma message("PROBE: NO async_to_lds path")
#endif

#define Bn   4
#define Nn   2048
#define FINn 128
#define Hn   4
#define HIDn 32
#define En   4
#define HFn  (Hn * HIDn) /* 128 */

typedef float v2f __attribute__((ext_vector_type(2)));
typedef float v4f __attribute__((ext_vector_type(4)));
typedef float v8f __attribute__((ext_vector_type(8)));
typedef int v4i_t __attribute__((vector_size(16))); // matches builtin signature
#define AS1 __attribute__((address_space(1)))
#define AS3 __attribute__((address_space(3)))

// ============================================================================
// Kernel 1: ht[b,h,i,d] = (h @ W) stored as (B,H,N,HID).
// One wave per 16x16 output tile; 32 K-steps of V_WMMA_F32_16X16X4_F32.
// ============================================================================
#if HAVE_WMMA_F32K4
__global__ __launch_bounds__(128) void k1_gemm_ht(const float* __restrict__ h,
                                                  const float* __restrict__ W,
                                                  float* __restrict__ ht) {
  const int wave = threadIdx.x >> 5;
  const int lane = threadIdx.x & 31;
  const int l16 = lane & 15;
  const int lhi = lane >> 4;
  const int tile = blockIdx.x * 4 + wave; // 0..4095
  const int m0 = (tile >> 3) * 16;
  const int n0 = (tile & 7) * 16;

  v8f c = {};
  // A 16x4 f32: lanes 0-15 hold K=0,1 in v0,v1; lanes 16-31 hold K=2,3
  const float* arow = h + (size_t)(m0 + l16) * FINn + lhi * 2;
#pragma unroll 4
  for (int kk = 0; kk < FINn; kk += 4) {
    v2f av = *(const v2f*)(arow + kk);
    const int br = kk + lhi * 2;
    v2f bv;
    bv.x = W[br * HFn + n0 + l16];
    bv.y = W[(br + 1) * HFn + n0 + l16];
    c = __builtin_amdgcn_wmma_f32_16x16x4_f32(false, av, false, bv, (short)0, c,
                                              false, false);
  }
#pragma unroll
  for (int r = 0; r < 8; ++r) {
    const int g = m0 + r + lhi * 8;
    const int b = g >> 11, i = g & (Nn - 1);
    const int col = n0 + l16;
    const int hd = col >> 5, d = col & 31;
    ht[(((size_t)(b * Hn + hd)) * Nn + i) * HIDn + d] = c[r];
  }
}
#else
__global__ __launch_bounds__(256) void k1_gemm_ht(const float* __restrict__ h,
                                                  const float* __restrict__ W,
                                                  float* __restrict__ ht) {
  const int t = blockIdx.x * blockDim.x + threadIdx.x;
  if (t >= Bn * Nn * HFn) return;
  const int g = t / HFn, col = t % HFn;
  const int b = g >> 11, i = g & (Nn - 1);
  const int hd = col >> 5, d = col & 31;
  float acc = 0.f;
  const float* row = h + (size_t)g * FINn;
  for (int k = 0; k < FINn; ++k) acc += row[k] * W[k * HFn + col];
  ht[(((size_t)(b * Hn + hd)) * Nn + i) * HIDn + d] = acc;
}
#endif

// ============================================================================
// Kernel 2: s_i[b,h,i] = ht . a_i[h],  s_j[b,h,i] = ht . a_j[h]
// ============================================================================
__global__ void k2_scores(const float* __restrict__ ht, const float* __restrict__ a,
                          float* __restrict__ si, float* __restrict__ sj) {
  const int t = blockIdx.x * blockDim.x + threadIdx.x; // ((b*H+h)*N + i)
  if (t >= Bn * Hn * Nn) return;
  const int hh = (t / Nn) % Hn;
  const float* row = ht + (size_t)t * HIDn;
  const float* ai = a + hh * (2 * HIDn + En);
  const float* aj = ai + HIDn;
  float aI = 0.f, aJ = 0.f;
#pragma unroll 8
  for (int d = 0; d < HIDn; ++d) {
    const float v = row[d];
    aI += v * ai[d];
    aJ += v * aj[d];
  }
  si[t] = aI;
  sj[t] = aJ;
}

// ============================================================================
// Kernel 3: Z[b,h,j] = sum_i mask*exp(beta), cnt[b,j] = #edges in column j
// ============================================================================
__global__ void k3_init(float* __restrict__ Z, int* __restrict__ cnt) {
  const int t = blockIdx.x * blockDim.x + threadIdx.x;
  if (t < Bn * Hn * Nn) Z[t] = 0.f;
  if (t < Bn * Nn) cnt[t] = 0;
}

__global__ __launch_bounds__(256) void k3_colsum(
    const float* __restrict__ ea, const int* __restrict__ adj,
    const float* __restrict__ si, const float* __restrict__ sj,
    const float* __restrict__ a, float* __restrict__ Z, int* __restrict__ cnt) {
  const int b = blockIdx.x;
  const int j = blockIdx.y * 256 + threadIdx.x;
  const int iBase = blockIdx.z * 256;

  float ae[Hn][En];
#pragma unroll
  for (int hh = 0; hh < Hn; ++hh)
#pragma unroll
    for (int e = 0; e < En; ++e) ae[hh][e] = a[hh * (2 * HIDn + En) + 2 * HIDn + e];

  float sjl[Hn];
#pragma unroll
  for (int hh = 0; hh < Hn; ++hh) sjl[hh] = sj[(b * Hn + hh) * Nn + j];

  float Zl[Hn] = {0.f, 0.f, 0.f, 0.f};
  int cl = 0;
  for (int i = iBase; i < iBase + 256; ++i) {
    // streaming data: non-temporal so the 335MB stream doesn't evict L2
    const v4f e4 = __builtin_nontemporal_load(
        (const v4f*)(ea + ((size_t)(b * Nn + i) * Nn + j) * En));
    const int m =
        __builtin_nontemporal_load(adj + (size_t)(b * Nn + i) * Nn + j);
    const float keep = (m > 0) ? 1.f : 0.f;
    cl += (m > 0) ? 1 : 0;
#pragma unroll
    for (int hh = 0; hh < Hn; ++hh) {
      const float siv = si[(b * Hn + hh) * Nn + i]; // uniform -> scalar load
      const float se = e4.x * ae[hh][0] + e4.y * ae[hh][1] + e4.z * ae[hh][2] +
                       e4.w * ae[hh][3];
      Zl[hh] += keep * __expf(siv + sjl[hh] + se);
    }
  }
#pragma unroll
  for (int hh = 0; hh < Hn; ++hh) atomicAdd(&Z[(b * Hn + hh) * Nn + j], Zl[hh]);
  atomicAdd(&cnt[b * Nn + j], cl);
}

__global__ void k3_fix(const float* __restrict__ Z, const int* __restrict__ cnt,
                       float* __restrict__ iz, float* __restrict__ fb) {
  const int t = blockIdx.x * blockDim.x + threadIdx.x;
  if (t >= Bn * Hn * Nn) return;
  const int b = t / (Hn * Nn);
  const int hh = (t / Nn) % Hn;
  const int j = t % Nn;
  const int c = cnt[b * Nn + j];
  iz[t] = (c == 0) ? 0.f : 1.f / Z[t];
  if (hh == 0) fb[b * Nn + j] = (c == 0) ? (1.f / (float)Nn) : 0.f;
}

// ============================================================================
// Kernel 4: h_prime[b,i,h*32+d] += sum_{j in split} P[i,j] * ht[b,h,j,d]
// P[i,j] = mask ? exp(s_i+s_j+s_e)*invZ[j] : fb[j]
// 4 waves = 4 heads share each (b, 16-row i-block, j-split); the j-stream is
// double-buffered through LDS with GLOBAL_LOAD_ASYNC_TO_LDS (ASYNCcnt); each
// wave does 4 K-steps x 2 V_WMMA_F32_16X16X4_F32 per 16-j chunk. Partial sums
// are combined across splits with global_atomic_add_f32.
// ============================================================================
#define JT 16
#define EAP 17 /* padded row stride to kill LDS bank conflicts */
#define JSPLIT 4
#define JRANGE (Nn / JSPLIT) /* 512 */
#define NCHUNK (JRANGE / JT) /* 32 */

__global__ void k4_init(float* __restrict__ out) {
  const int t = blockIdx.x * blockDim.x + threadIdx.x;
  if (t < Bn * Nn * HFn) out[t] = 0.f;
}

__global__ __launch_bounds__(128) void k4_attn_out(
    const float* __restrict__ ea, const int* __restrict__ adj,
    const float* __restrict__ ht, const float* __restrict__ si,
    const float* __restrict__ sj, const float* __restrict__ iz,
    const float* __restrict__ fb, const float* __restrict__ avec,
    float* __restrict__ out) {
  __shared__ __align__(16) float ea_sh[2][16 * EAP * En];
  __shared__ __align__(16) int adj_sh[2][16 * EAP];
  __shared__ float sj_sh[2][Hn][JT];
  __shared__ float iz_sh[2][Hn][JT];
  __shared__ float fb_sh[2][JT];
  __shared__ float si_sh[Hn][16];

  const int wg = blockIdx.x; // Bn * (Nn/16) * JSPLIT
  const int split = wg % JSPLIT;
  const int ib = wg / JSPLIT;
  const int b = ib / (Nn / 16);
  const int i0 = (ib % (Nn / 16)) * 16;
  const int jstart = split * JRANGE;

  const int tid = threadIdx.x;
  const int wave = tid >> 5; // head
  const int lane = tid & 31;
  const int l16 = lane & 15;
  const int lhi = lane >> 4;

  const float ae0 = avec[wave * (2 * HIDn + En) + 2 * HIDn + 0];
  const float ae1 = avec[wave * (2 * HIDn + En) + 2 * HIDn + 1];
  const float ae2 = avec[wave * (2 * HIDn + En) + 2 * HIDn + 2];
  const float ae3 = avec[wave * (2 * HIDn + En) + 2 * HIDn + 3];

  if (tid < Hn * 16) {
    const int hh = tid >> 4, ii = tid & 15;
    si_sh[hh][ii] = si[(b * Hn + hh) * Nn + i0 + ii];
  }

  // stage one 16x16 (i x j) tile of edge_attr/adj + per-j stats into buffer
  auto stage = [&](int buf, int j0) {
#pragma unroll
    for (int r = 0; r < 2; ++r) {
      const int idx = tid + r * 128; // 0..255 -> (i,j) cell
      const int ii = idx >> 4, jj = idx & 15;
      const float* gsrc = ea + ((size_t)(b * Nn + i0 + ii) * Nn + j0 + jj) * En;
      const int* asrc = adj + (size_t)(b * Nn + i0 + ii) * Nn + j0 + jj;
#if HAVE_ASYNC
      __builtin_amdgcn_global_load_async_to_lds_b128(
          (AS1 v4i_t*)gsrc, (AS3 v4i_t*)&ea_sh[buf][(ii * EAP + jj) * En], 0, 0);
      __builtin_amdgcn_global_load_async_to_lds_b32(
          (AS1 int*)asrc, (AS3 int*)&adj_sh[buf][ii * EAP + jj], 0, 0);
#else
      *(v4f*)&ea_sh[buf][(ii * EAP + jj) * En] =
          __builtin_nontemporal_load((const v4f*)gsrc);
      adj_sh[buf][ii * EAP + jj] = __builtin_nontemporal_load(asrc);
#endif
    }
    if (tid < Hn * JT) {
      const int hh = tid >> 4, jj = tid & 15;
      sj_sh[buf][hh][jj] = sj[(b * Hn + hh) * Nn + j0 + jj];
      iz_sh[buf][hh][jj] = iz[(b * Hn + hh) * Nn + j0 + jj];
    }
    if (tid < JT) fb_sh[buf][tid] = fb[b * Nn + j0 + tid];
  };

#if HAVE_WMMA_F32K4
  v8f c0 = {};
  v8f c1 = {};
#else
  float acc[16];
#pragma unroll
  for (int r = 0; r < 16; ++r) acc[r] = 0.f;
#endif

  const float* htb = ht + (size_t)(b * Hn + wave) * Nn * HIDn;

  stage(0, jstart);
  for (int jc = 0; jc < NCHUNK; ++jc) {
    const int cur = jc & 1;
    const int j0 = jstart + jc * JT;
    if (jc + 1 < NCHUNK) stage(cur ^ 1, j0 + JT); // prefetch next chunk
#if HAVE_ASYNC
    // each stage = 4 async ops/wave; <=4 outstanding means current buffer done
    if (jc + 1 < NCHUNK) __builtin_amdgcn_s_wait_asynccnt(4);
    else                 __builtin_amdgcn_s_wait_asynccnt(0);
#endif
    __syncthreads(); // current buffer visible to all waves

#if HAVE_WMMA_F32K4
#pragma unroll
    for (int ks = 0; ks < 4; ++ks) {
      const int jb = ks * 4;
      // A-frag: lane covers (i = l16, j = jb + lhi*2 + {0,1})
      const int ii = l16;
      const float siv = si_sh[wave][ii];
      v2f af;
      {
        const int jj = jb + lhi * 2;
        const v4f e4 = *(const v4f*)&ea_sh[cur][(ii * EAP + jj) * En];
        const float se = e4.x * ae0 + e4.y * ae1 + e4.z * ae2 + e4.w * ae3;
        const float pe =
            __expf(siv + sj_sh[cur][wave][jj] + se) * iz_sh[cur][wave][jj];
        af.x = (adj_sh[cur][ii * EAP + jj] > 0) ? pe : fb_sh[cur][jj];
      }
      {
        const int jj = jb + lhi * 2 + 1;
        const v4f e4 = *(const v4f*)&ea_sh[cur][(ii * EAP + jj) * En];
        const float se = e4.x * ae0 + e4.y * ae1 + e4.z * ae2 + e4.w * ae3;
        const float pe =
            __expf(siv + sj_sh[cur][wave][jj] + se) * iz_sh[cur][wave][jj];
        af.y = (adj_sh[cur][ii * EAP + jj] > 0) ? pe : fb_sh[cur][jj];
      }
      // B-frags from ht: K rows j, 16 d columns; lanes 0-15 K={0,1}, 16-31 K={2,3}
      const int r0 = j0 + jb + lhi * 2;
      v2f b0, b1;
      b0.x = htb[(size_t)r0 * HIDn + l16];
      b0.y = htb[(size_t)(r0 + 1) * HIDn + l16];
      b1.x = htb[(size_t)r0 * HIDn + 16 + l16];
      b1.y = htb[(size_t)(r0 + 1) * HIDn + 16 + l16];
      c0 = __builtin_amdgcn_wmma_f32_16x16x4_f32(false, af, false, b0, (short)0,
                                                 c0, false, false);
      c1 = __builtin_amdgcn_wmma_f32_16x16x4_f32(false, af, false, b1, (short)0,
                                                 c1, false, false);
    }
#else
    for (int jj = 0; jj < JT; ++jj) {
      const float htv = htb[(size_t)(j0 + jj) * HIDn + lane];
      const float sjv = sj_sh[cur][wave][jj];
      const float izv = iz_sh[cur][wave][jj];
      const float fbv = fb_sh[cur][jj];
#pragma unroll
      for (int ii = 0; ii < 16; ++ii) {
        const v4f e4 = *(const v4f*)&ea_sh[cur][(ii * EAP + jj) * En];
        const float se = e4.x * ae0 + e4.y * ae1 + e4.z * ae2 + e4.w * ae3;
        const float pe = __expf(si_sh[wave][ii] + sjv + se) * izv;
        const float p = (adj_sh[cur][ii * EAP + jj] > 0) ? pe : fbv;
        acc[ii] += p * htv;
      }
    }
#endif
    __syncthreads(); // all waves done with buffer before it is re-staged
  }

  // combine split partial sums
#if HAVE_WMMA_F32K4
#pragma unroll
  for (int r = 0; r < 8; ++r) {
    const int row = i0 + r + lhi * 8;
    float* o = out + (size_t)(b * Nn + row) * HFn + wave * HIDn;
    atomicAdd(&o[l16], c0[r]);
    atomicAdd(&o[16 + l16], c1[r]);
  }
#else
#pragma unroll
  for (int ii = 0; ii < 16; ++ii)
    atomicAdd(&out[(size_t)(b * Nn + i0 + ii) * HFn + wave * HIDn + lane],
              acc[ii]);
#endif
}

// ============================================================================
extern "C" void kernel_launch(void* const* d_in, const int* in_sizes, int n_in,
                              void* d_out, int out_size, void* d_ws,
                              size_t ws_size, hipStream_t stream) {
  const float* h = (const float*)d_in[0];
  const int* adj = (const int*)d_in[1];
  const float* ea = (const float*)d_in[2];
  const float* W = (const float*)d_in[3];
  const float* a = (const float*)d_in[4];
  float* out = (float*)d_out;

  float* ht_ws = (float*)d_ws;                        // B*H*N*HID = 1048576
  float* si_ws = ht_ws + (size_t)Bn * Hn * Nn * HIDn; // 32768
  float* sj_ws = si_ws + Bn * Hn * Nn;                // 32768
  float* Z_ws = sj_ws + Bn * Hn * Nn;                 // 32768
  float* iz_ws = Z_ws + Bn * Hn * Nn;                 // 32768
  float* fb_ws = iz_ws + Bn * Hn * Nn;                // B*N = 8192
  int* cnt_ws = (int*)(fb_ws + Bn * Nn);              // 8192 ints

  // k1: ht = h @ W
#if HAVE_WMMA_F32K4
  k1_gemm_ht<<<(Bn * Nn / 16) * (HFn / 16) / 4, 128, 0, stream>>>(h, W, ht_ws);
#else
  k1_gemm_ht<<<(Bn * Nn * HFn + 255) / 256, 256, 0, stream>>>(h, W, ht_ws);
#endif

  // k2: node scores
  k2_scores<<<(Bn * Hn * Nn) / 256, 256, 0, stream>>>(ht_ws, a, si_ws, sj_ws);

  // k3: column softmax denominators
  k3_init<<<(Bn * Hn * Nn) / 256, 256, 0, stream>>>(Z_ws, cnt_ws);
  dim3 g3(Bn, Nn / 256, Nn / 256);
  k3_colsum<<<g3, 256, 0, stream>>>(ea, adj, si_ws, sj_ws, a, Z_ws, cnt_ws);
  k3_fix<<<(Bn * Hn * Nn) / 256, 256, 0, stream>>>(Z_ws, cnt_ws, iz_ws, fb_ws);

  // k4: attention-weighted aggregation
  k4_init<<<(Bn * Nn * HFn) / 256, 256, 0, stream>>>(out);
  k4_attn_out<<<Bn * (Nn / 16) * JSPLIT, 128, 0, stream>>>(
      ea, adj, ht_ws, si_ws, sj_ws, iz_ws, fb_ws, a, out);
}